// Comb1d_46737834115366
// MI455X (gfx1250) — compile-verified
//
#include <hip/hip_runtime.h>

// ---- problem constants (from reference) ------------------------------------
#define SRATE_F   16000.0f
#define NTAPS     10
#define T_LEN     32000
#define B_N       8
#define O_N       128
#define MAXD      2880            // ceil(9 * 16000/50) max integer delay

// ---- tiling ----------------------------------------------------------------
#define TILE      1024            // output samples per block
#define OG        16              // output channels per block
#define THREADS   256             // 8 wave32 waves
#define WIN_PAD   4096            // (TILE + MAXD + 1 = 3905) padded to 4-dword*256

// CDNA5 async global->LDS copies (tracked by ASYNCcnt). The LDS destination
// address is the low 32 bits of the flat shared pointer (HW adds LDS_BASE).
__device__ __forceinline__ void async_lds_b128(unsigned lds_off,
                                               unsigned long long gaddr) {
    asm volatile("global_load_async_to_lds_b128 %0, %1, off"
                 :: "v"(lds_off), "v"(gaddr) : "memory");
}
__device__ __forceinline__ void async_lds_b32(unsigned lds_off,
                                              unsigned long long gaddr) {
    asm volatile("global_load_async_to_lds_b32 %0, %1, off"
                 :: "v"(lds_off), "v"(gaddr) : "memory");
}
__device__ __forceinline__ void wait_async0() {
    asm volatile("s_wait_asynccnt 0" ::: "memory");
}

__global__ __launch_bounds__(THREADS)
void comb1d_kernel(const float* __restrict__ x,
                   const float* __restrict__ f,
                   const float* __restrict__ a,
                   float* __restrict__ out)
{
    __shared__ __align__(16) float s_x[WIN_PAD];   // 16 KB window of x[b]
    __shared__ float s_kd[OG * NTAPS];             // k * D[o]  (fp32, as ref)
    __shared__ float s_ak[OG * NTAPS];             // a[o]^k

    const int tile = blockIdx.x;
    const int og   = blockIdx.y;
    const int b    = blockIdx.z;
    const int tid  = threadIdx.x;

    const int t0     = tile * TILE;
    const int win_lo = (t0 - MAXD) > 0 ? (t0 - MAXD) : 0;   // multiple of 4
    int win_hi = t0 + TILE;
    if (win_hi > T_LEN - 1) win_hi = T_LEN - 1;
    const int cnt = win_hi - win_lo + 1;                    // <= 3905

    const float* xg = x + (size_t)b * T_LEN;

    // ---- 1) kick off async staging of the x window into LDS ----------------
    // Issued FIRST so the DMA engine overlaps the transcendental weight math.
    {
        const int n_grp     = (cnt + 3) >> 2;                         // <= 977
        const int n_grp_pad = (n_grp + THREADS - 1) & ~(THREADS - 1); // <= 1024
        for (int g = tid; g < n_grp_pad; g += THREADS) {
            const int gbase = win_lo + 4 * g;                 // dword index
            if (gbase + 4 <= T_LEN) {                         // 16B-aligned
                async_lds_b128((unsigned)(size_t)&s_x[4 * g],
                               (unsigned long long)(size_t)(xg + gbase));
            } else {                                          // tensor tail
                #pragma unroll
                for (int j = 0; j < 4; ++j) {
                    int gi = gbase + j;
                    if (gi > T_LEN - 1) gi = T_LEN - 1;
                    async_lds_b32((unsigned)(size_t)&s_x[4 * g + j],
                                  (unsigned long long)(size_t)(xg + gi));
                }
            }
        }
    }

    // ---- 2) per-(channel, tap) constants (overlaps async DMA) --------------
    if (tid < OG * NTAPS) {
        const int ol = tid / NTAPS;
        const int k  = tid % NTAPS;
        const int o  = og * OG + ol;
        const float fr = f[o];
        const float sg = 1.0f / (1.0f + expf(-fr));               // sigmoid
        const float fo = 50.0f * expf(sg * 3.6888794541139363f);  // 50*40^sg
        const float D  = SRATE_F / fo;
        s_kd[tid] = (float)k * D;
        const float av = a[o];
        float ak = 1.0f;
        for (int j = 0; j < k; ++j) ak *= av;                     // a^k
        s_ak[tid] = ak;
    }

    // ---- 3) wait for DMA, publish to all waves ------------------------------
    wait_async0();
    __syncthreads();

    const int o_base = og * OG;
    // Interior tiles: pos >= t0-2880 >= 0 (no valid mask / low clamp) and
    // i1 <= t0+TILE <= T-1 (no high clamp). Covers 28 of 32 tiles.
    const bool fast = (t0 >= MAXD) && (t0 + TILE <= T_LEN - 1);

    if (fast) {
        for (int ol = 0; ol < OG; ++ol) {
            float kd[NTAPS], ak[NTAPS];
            #pragma unroll
            for (int k = 0; k < NTAPS; ++k) {
                kd[k] = s_kd[ol * NTAPS + k];
                ak[k] = s_ak[ol * NTAPS + k];
            }
            float* orow = out + ((size_t)b * O_N + (o_base + ol)) * T_LEN;
            #pragma unroll
            for (int tt = 0; tt < TILE / THREADS; ++tt) {
                const int t  = t0 + tt * THREADS + tid;
                const float tf = (float)t;
                float acc = 0.0f;
                #pragma unroll
                for (int k = 0; k < NTAPS; ++k) {
                    const float pos  = tf - kd[k];     // fp32, as reference
                    const float i0f  = floorf(pos);
                    const float frac = pos - i0f;
                    const int   li   = (int)i0f - win_lo;
                    const float x0 = s_x[li];          // merges to 2ADDR load
                    const float x1 = s_x[li + 1];
                    acc = fmaf(ak[k], fmaf(frac, x1 - x0, x0), acc);
                }
                orow[t] = acc;
            }
        }
    } else {
        const int t_end = (t0 + TILE) < T_LEN ? (t0 + TILE) : T_LEN;
        for (int ol = 0; ol < OG; ++ol) {
            float kd[NTAPS], ak[NTAPS];
            #pragma unroll
            for (int k = 0; k < NTAPS; ++k) {
                kd[k] = s_kd[ol * NTAPS + k];
                ak[k] = s_ak[ol * NTAPS + k];
            }
            float* orow = out + ((size_t)b * O_N + (o_base + ol)) * T_LEN;
            for (int t = t0 + tid; t < t_end; t += THREADS) {
                const float tf = (float)t;
                float acc = 0.0f;
                #pragma unroll
                for (int k = 0; k < NTAPS; ++k) {
                    const float pos   = tf - kd[k];
                    const float i0f   = floorf(pos);
                    const float frac  = pos - i0f;
                    const float valid = (i0f >= 0.0f) ? 1.0f : 0.0f;
                    int ic = (int)i0f;
                    if (ic < 0) ic = 0;                    // ref low clamp
                    int i1 = ic + 1;
                    if (i1 > T_LEN - 1) i1 = T_LEN - 1;    // ref high clamp
                    const float x0 = s_x[ic - win_lo];
                    const float x1 = s_x[i1 - win_lo];
                    const float tap = fmaf(frac, x1 - x0, x0) * valid;
                    acc = fmaf(ak[k], tap, acc);
                }
                orow[t] = acc;
            }
        }
    }
}

extern "C" void kernel_launch(void* const* d_in, const int* in_sizes, int n_in,
                              void* d_out, int out_size, void* d_ws, size_t ws_size,
                              hipStream_t stream) {
    (void)in_sizes; (void)n_in; (void)out_size; (void)d_ws; (void)ws_size;
    const float* x = (const float*)d_in[0];   // (8, 1, 32000)
    const float* f = (const float*)d_in[1];   // (128, 1)
    const float* a = (const float*)d_in[2];   // (128, 1)
    float* out = (float*)d_out;               // (8, 128, 32000)

    dim3 grid((T_LEN + TILE - 1) / TILE, O_N / OG, B_N);  // (32, 8, 8)
    comb1d_kernel<<<grid, THREADS, 0, stream>>>(x, f, a, out);
}